// PointTransformerTransitionDown_5617817224083
// MI455X (gfx1250) — compile-verified
//
#include <hip/hip_runtime.h>

#define BATCH 8
#define NPTS  8192
#define SPTS  2048
#define KNB   16
#define INC   64
#define OUTC  128
#define FAN   67
#define FANP  68

typedef float v2f __attribute__((ext_vector_type(2)));
typedef float v8f __attribute__((ext_vector_type(8)));

// ------------------------------------------------------------------
// Farthest point sampling: one 1024-thread block per batch.
// dist[] lives in LDS (32KB); argmax via shared-memory tree reduction
// with first-index tie-break (matches jnp.argmax).
// Also writes new_xyz (first chunk of d_out) and centers (ws).
// ------------------------------------------------------------------
__global__ __launch_bounds__(1024) void fps_kernel(const float* __restrict__ xyz,
                                                   int* __restrict__ fps_idx,
                                                   float* __restrict__ centers,
                                                   float* __restrict__ new_xyz_out) {
  __shared__ float dist[NPTS];
  __shared__ float rv[1024];
  __shared__ int   ri[1024];
  __shared__ int   sfar;
  const int b = blockIdx.x;
  const int t = threadIdx.x;
  const float* xb = xyz + (size_t)b * 3 * NPTS;

#pragma unroll
  for (int i = 0; i < 8; ++i) dist[t * 8 + i] = 1e10f;
  if (t == 0) sfar = 0;
  __syncthreads();

  for (int s = 0; s < SPTS; ++s) {
    const int far = sfar;
    if (t == 0) {
      fps_idx[b * SPTS + s] = far;
      float x = xb[far], y = xb[NPTS + far], z = xb[2 * NPTS + far];
      new_xyz_out[(size_t)b * 3 * SPTS + 0 * SPTS + s] = x;
      new_xyz_out[(size_t)b * 3 * SPTS + 1 * SPTS + s] = y;
      new_xyz_out[(size_t)b * 3 * SPTS + 2 * SPTS + s] = z;
      centers[((size_t)b * SPTS + s) * 3 + 0] = x;
      centers[((size_t)b * SPTS + s) * 3 + 1] = y;
      centers[((size_t)b * SPTS + s) * 3 + 2] = z;
    }
    const float cx = xb[far], cy = xb[NPTS + far], cz = xb[2 * NPTS + far];
    float bestv = -1.0f;
    int   besti = 0;
#pragma unroll
    for (int i = 0; i < 8; ++i) {
      const int n = t * 8 + i;
      float dx = xb[n] - cx, dy = xb[NPTS + n] - cy, dz = xb[2 * NPTS + n] - cz;
      float d  = dx * dx + dy * dy + dz * dz;
      float dn = fminf(dist[n], d);
      dist[n] = dn;
      if (dn > bestv) { bestv = dn; besti = n; }
    }
    rv[t] = bestv; ri[t] = besti;
    __syncthreads();
    for (int off = 512; off > 0; off >>= 1) {
      if (t < off) {
        float ov = rv[t + off]; int oi = ri[t + off];
        if (ov > rv[t] || (ov == rv[t] && oi < ri[t])) { rv[t] = ov; ri[t] = oi; }
      }
      __syncthreads();
    }
    if (t == 0) sfar = ri[0];
    __syncthreads();
  }
}

// ------------------------------------------------------------------
// Brute-force KNN: one thread per center, register insertion-sort of
// the K=16 smallest distances (strict < keeps earliest index on ties,
// matching top_k stability).
// ------------------------------------------------------------------
__global__ __launch_bounds__(256) void knn_kernel(const float* __restrict__ xyz,
                                                  const float* __restrict__ centers,
                                                  int* __restrict__ gidx) {
  const int g = blockIdx.x * blockDim.x + threadIdx.x;
  if (g >= BATCH * SPTS) return;
  const int b = g / SPTS;
  const float* xb = xyz + (size_t)b * 3 * NPTS;
  const float cx = centers[g * 3 + 0], cy = centers[g * 3 + 1], cz = centers[g * 3 + 2];

  float bd[KNB]; int bi[KNB];
#pragma unroll
  for (int j = 0; j < KNB; ++j) { bd[j] = 1e30f; bi[j] = 0; }

  for (int n = 0; n < NPTS; ++n) {
    float dx = xb[n] - cx, dy = xb[NPTS + n] - cy, dz = xb[2 * NPTS + n] - cz;
    float d = dx * dx + dy * dy + dz * dz;
    if (d < bd[KNB - 1]) {
      int pos = KNB - 1;
#pragma unroll
      for (int j = KNB - 2; j >= 0; --j) {
        if (d < bd[j]) { bd[j + 1] = bd[j]; bi[j + 1] = bi[j]; pos = j; }
      }
      bd[pos] = d; bi[pos] = n;
    }
  }
#pragma unroll
  for (int j = 0; j < KNB; ++j) gidx[(size_t)g * KNB + j] = bi[j];
}

__global__ void zero_stats(float* __restrict__ stats) {
  if (threadIdx.x < 2 * OUTC) stats[threadIdx.x] = 0.0f;
}

__global__ void finalize_stats(const float* __restrict__ stats,
                               const float* __restrict__ gamma,
                               const float* __restrict__ beta,
                               float* __restrict__ scsh) {
  const int o = threadIdx.x;
  if (o < OUTC) {
    const float cnt  = (float)BATCH * (float)SPTS * (float)KNB;
    float mean = stats[o] / cnt;
    float var  = stats[OUTC + o] / cnt - mean * mean;
    float a    = rsqrtf(var + 1e-5f) * gamma[o];
    scsh[o]        = a;
    scsh[OUTC + o] = beta[o] - mean * a;
  }
}

// grouped[c][neighbor] element on the fly (points channel / relative xyz / pad)
__device__ __forceinline__ float fetch_grouped(const float* __restrict__ pb,
                                               const float* __restrict__ xyzb,
                                               int c, int pidx,
                                               float cx, float cy, float cz) {
  if (c < INC)     return pb[(size_t)c * NPTS + pidx];
  if (c == INC)     return xyzb[pidx] - cx;
  if (c == INC + 1) return xyzb[NPTS + pidx] - cy;
  if (c == INC + 2) return xyzb[2 * NPTS + pidx] - cz;
  return 0.0f;
}

// ------------------------------------------------------------------
// Fused 1x1-conv GEMM on V_WMMA_F32_16X16X4_F32.
// Each wave owns one (b,s): its 16-column N-tile is exactly the 16
// neighbors. M=128 -> 8 accumulator tiles, K=67 padded to 68 -> 17 steps.
// mode 0: accumulate per-channel sum/sumsq (LDS atomics -> global).
// mode 1: affine-BN + ReLU, then K-max via 16-lane shfl_xor reduction.
// ------------------------------------------------------------------
__global__ __launch_bounds__(256) void conv_bn_kernel(
    const float* __restrict__ xyz, const float* __restrict__ points,
    const float* __restrict__ w,   const float* __restrict__ bias,
    const int*   __restrict__ gidx, const float* __restrict__ centers,
    float* __restrict__ stats, const float* __restrict__ scsh,
    float* __restrict__ outp, int mode) {
  __shared__ float wLDS[OUTC * FANP];
  __shared__ float statsLDS[2 * OUTC];

  for (int i = threadIdx.x; i < OUTC * FANP; i += 256) {
    const int o = i / FANP, c = i % FANP;
    wLDS[i] = (c < FAN) ? w[o * FAN + c] : 0.0f;
  }
  if (mode == 0 && threadIdx.x < 2 * OUTC) statsLDS[threadIdx.x] = 0.0f;
  __syncthreads();

  const int wave = threadIdx.x >> 5, lane = threadIdx.x & 31;
  const int b = blockIdx.x / (SPTS / 8);
  const int s = (blockIdx.x % (SPTS / 8)) * 8 + wave;
  const int half = lane >> 4, rin = lane & 15;
  const int g = b * SPTS + s;
  const int pidx = gidx[(size_t)g * KNB + rin];   // column = neighbor rin
  const float cx = centers[g * 3 + 0], cy = centers[g * 3 + 1], cz = centers[g * 3 + 2];
  const float* pb   = points + (size_t)b * INC * NPTS;
  const float* xyzb = xyz    + (size_t)b * 3 * NPTS;

  v8f acc[8];
  const v8f vzero = {0.f, 0.f, 0.f, 0.f, 0.f, 0.f, 0.f, 0.f};
#pragma unroll
  for (int mt = 0; mt < 8; ++mt) acc[mt] = vzero;

  for (int kc = 0; kc < 17; ++kc) {
    const int c0 = 4 * kc + 2 * half;   // lanes 0-15: K=0,1 ; lanes 16-31: K=2,3
    v2f bf;
    bf.x = fetch_grouped(pb, xyzb, c0,     pidx, cx, cy, cz);
    bf.y = fetch_grouped(pb, xyzb, c0 + 1, pidx, cx, cy, cz);
#pragma unroll
    for (int mt = 0; mt < 8; ++mt) {
      v2f af;
      af.x = wLDS[(16 * mt + rin) * FANP + c0];
      af.y = wLDS[(16 * mt + rin) * FANP + c0 + 1];
      acc[mt] = __builtin_amdgcn_wmma_f32_16x16x4_f32(
          false, af, false, bf, (short)0, acc[mt], false, false);
    }
  }

  if (mode == 0) {
#pragma unroll
    for (int mt = 0; mt < 8; ++mt) {
#pragma unroll
      for (int r = 0; r < 8; ++r) {
        const int o = 16 * mt + r + 8 * half;   // C-layout: vgpr r = rows r / r+8
        const float val = acc[mt][r] + bias[o];
        atomicAdd(&statsLDS[o], val);
        atomicAdd(&statsLDS[OUTC + o], val * val);
      }
    }
    __syncthreads();
    if (threadIdx.x < 2 * OUTC) atomicAdd(&stats[threadIdx.x], statsLDS[threadIdx.x]);
  } else {
#pragma unroll
    for (int mt = 0; mt < 8; ++mt) {
#pragma unroll
      for (int r = 0; r < 8; ++r) {
        const int o = 16 * mt + r + 8 * half;
        float val = acc[mt][r] + bias[o];
        val = fmaxf(val * scsh[o] + scsh[OUTC + o], 0.0f);
        // max over the 16 neighbor columns: xor masks 1..8 stay inside each
        // 16-lane half of the wave32.
#pragma unroll
        for (int m = 1; m < 16; m <<= 1) val = fmaxf(val, __shfl_xor(val, m, 32));
        if (rin == 0) outp[((size_t)b * OUTC + o) * SPTS + s] = val;
      }
    }
  }
}

extern "C" void kernel_launch(void* const* d_in, const int* in_sizes, int n_in,
                              void* d_out, int out_size, void* d_ws, size_t ws_size,
                              hipStream_t stream) {
  const float* xyz    = (const float*)d_in[0];
  const float* points = (const float*)d_in[1];
  const float* w      = (const float*)d_in[2];
  const float* bias   = (const float*)d_in[3];
  const float* gamma  = (const float*)d_in[4];
  const float* beta   = (const float*)d_in[5];

  float* out        = (float*)d_out;
  float* new_xyz    = out;                       // [B,3,S]
  float* new_points = out + (size_t)BATCH * 3 * SPTS;  // [B,OUTC,S]

  char* ws = (char*)d_ws;
  int*   fps_idx = (int*)ws;                                  //  64 KB
  float* centers = (float*)(ws + 65536);                      // 192 KB
  int*   gidx    = (int*)(ws + 65536 + 196608);               //   1 MB
  float* stats   = (float*)(ws + 65536 + 196608 + 1048576);   //   1 KB
  float* scsh    = (float*)(ws + 65536 + 196608 + 1048576 + 1024);

  fps_kernel<<<BATCH, 1024, 0, stream>>>(xyz, fps_idx, centers, new_xyz);
  knn_kernel<<<(BATCH * SPTS) / 256, 256, 0, stream>>>(xyz, centers, gidx);
  zero_stats<<<1, 256, 0, stream>>>(stats);
  conv_bn_kernel<<<(BATCH * SPTS) / 8, 256, 0, stream>>>(
      xyz, points, w, bias, gidx, centers, stats, scsh, new_points, 0);
  finalize_stats<<<1, 128, 0, stream>>>(stats, gamma, beta, scsh);
  conv_bn_kernel<<<(BATCH * SPTS) / 8, 256, 0, stream>>>(
      xyz, points, w, bias, gidx, centers, stats, scsh, new_points, 1);
}